// Autoformer_34823594836232
// MI455X (gfx1250) — compile-verified
//
#include <hip/hip_runtime.h>
#include <math.h>

// ---------------------------------------------------------------------------
// Autoformer forward for MI455X (gfx1250, wave32).
// GEMMs use V_WMMA_F32_16X16X4_F32 (fp32 matrix path), 32x32 per wave
// (4 accumulators) for 2 flt/WMMA fragment traffic.
// Autocorrelation: sum Qf*conj(Kf) over channels in spectrum space (irfft is
// linear), one inverse FFT per batch -> mean_value directly.
// Workspace requirement: ~341 MB.
// ---------------------------------------------------------------------------

#define BB    8
#define LSEQ  2048
#define DM    512
#define MTOT  (BB * LSEQ)      // 16384 rows
#define TOPK  38               // int(5 * ln(2048))
#define NCHUNK 16              // FFT d-chunks (32 channels each)

typedef float v2f __attribute__((ext_vector_type(2)));
typedef float v8f __attribute__((ext_vector_type(8)));

__device__ __forceinline__ float gelu_f(float x) {
  return 0.5f * x * (1.0f + erff(x * 0.7071067811865475f));
}

// ============================ WMMA GEMM ====================================
// C[M,N] = A[M,K] @ W[K,N] (+bias) (gelu?) (+residual)
// Block tile 128x64, 8 waves in 4(M) x 2(N); each wave owns a 32x32 strip:
// 2 M-subtiles x 2 N-subtiles = 4 f32 16x16 accumulators. K stepped by 4.
#define BM 128
#define BN 64
#define BK 16

__global__ __launch_bounds__(256) void k_gemm_wmma(
    const float* __restrict__ A, const float* __restrict__ W,
    const float* __restrict__ bias, const float* __restrict__ residual,
    float* __restrict__ C, int M, int N, int K, int gelu_flag)
{
  __shared__ float As[BM][BK + 1];
  __shared__ float Bs[BK][BN + 1];
  const int bm = blockIdx.y * BM;
  const int bn = blockIdx.x * BN;
  const int tid  = threadIdx.x;
  const int wave = tid >> 5;        // 0..7 (wave32)
  const int lane = tid & 31;
  const int wm = wave & 3;          // M sub-strip (32 rows each)
  const int wn = wave >> 2;         // N sub-strip (32 cols each)
  const int lm = lane & 15;
  const int khalf = (lane >> 4) * 2; // lanes 0-15 -> K{0,1}; 16-31 -> K{2,3}

  v8f acc00 = {}, acc01 = {}, acc10 = {}, acc11 = {};

  for (int k0 = 0; k0 < K; k0 += BK) {
    for (int i = tid; i < BM * BK; i += 256) {
      int r = i >> 4, c = i & 15;
      As[r][c] = A[(size_t)(bm + r) * K + (k0 + c)];
    }
    for (int i = tid; i < BK * BN; i += 256) {
      int r = i >> 6, c = i & 63;
      Bs[r][c] = W[(size_t)(k0 + r) * N + (bn + c)];
    }
    if (k0 + BK < K) {  // prefetch next K-chunk (global_prefetch_b8)
      int r0 = tid >> 4, c0 = tid & 15;
      __builtin_prefetch(&A[(size_t)(bm + r0) * K + (k0 + BK) + c0], 0, 3);
      __builtin_prefetch(&A[(size_t)(bm + 64 + r0) * K + (k0 + BK) + c0], 0, 3);
      __builtin_prefetch(&W[(size_t)(k0 + BK + (tid >> 6)) * N + bn + (tid & 63)], 0, 3);
    }
    __syncthreads();
    const int mrow0 = wm * 32 + lm;
    const int mrow1 = mrow0 + 16;
    const int ncol  = wn * 32 + lm;
#pragma unroll
    for (int ks = 0; ks < BK; ks += 4) {
      v2f a0; a0.x = As[mrow0][ks + khalf];    a0.y = As[mrow0][ks + khalf + 1];
      v2f a1; a1.x = As[mrow1][ks + khalf];    a1.y = As[mrow1][ks + khalf + 1];
      v2f b0; b0.x = Bs[ks + khalf][ncol];     b0.y = Bs[ks + khalf + 1][ncol];
      v2f b1; b1.x = Bs[ks + khalf][ncol+16];  b1.y = Bs[ks + khalf + 1][ncol+16];
      acc00 = __builtin_amdgcn_wmma_f32_16x16x4_f32(false, a0, false, b0,
                                                    (short)0, acc00, false, false);
      acc01 = __builtin_amdgcn_wmma_f32_16x16x4_f32(false, a0, false, b1,
                                                    (short)0, acc01, false, false);
      acc10 = __builtin_amdgcn_wmma_f32_16x16x4_f32(false, a1, false, b0,
                                                    (short)0, acc10, false, false);
      acc11 = __builtin_amdgcn_wmma_f32_16x16x4_f32(false, a1, false, b1,
                                                    (short)0, acc11, false, false);
    }
    __syncthreads();
  }

  // C/D layout: VGPR r holds (M = r + 8*(lane>=16), N = lane%16)
  const int rb0  = bm + wm * 32 + (lane >> 4) * 8;      // M-subtile 0
  const int rb1  = rb0 + 16;                            // M-subtile 1
  const int col0 = bn + wn * 32 + lm;
  const int col1 = col0 + 16;
  float bias0 = 0.f, bias1 = 0.f;
  if (bias) { bias0 = bias[col0]; bias1 = bias[col1]; }
#pragma unroll
  for (int r = 0; r < 8; ++r) {
    int row0 = rb0 + r, row1 = rb1 + r;
    float v00 = acc00[r] + bias0, v01 = acc01[r] + bias1;
    float v10 = acc10[r] + bias0, v11 = acc11[r] + bias1;
    if (gelu_flag) {
      v00 = gelu_f(v00); v01 = gelu_f(v01);
      v10 = gelu_f(v10); v11 = gelu_f(v11);
    }
    if (residual) {
      v00 += residual[(size_t)row0 * N + col0];
      v01 += residual[(size_t)row0 * N + col1];
      v10 += residual[(size_t)row1 * N + col0];
      v11 += residual[(size_t)row1 * N + col1];
    }
    C[(size_t)row0 * N + col0] = v00;
    C[(size_t)row0 * N + col1] = v01;
    C[(size_t)row1 * N + col0] = v10;
    C[(size_t)row1 * N + col1] = v11;
  }
}

// ============================ FFT autocorrelation ==========================
#define FFT_N   2048
#define FFT_LOG 11

__device__ void fft_lds(float2* data, int tid, float sign) {
  for (int s = 1; s <= FFT_LOG; ++s) {
    const int m = 1 << s, half = m >> 1;
    for (int j = tid; j < FFT_N / 2; j += 256) {
      int grp = j >> (s - 1);
      int pos = j & (half - 1);
      int i1 = grp * m + pos;
      int i2 = i1 + half;
      float ang = sign * 6.283185307179586f * (float)pos / (float)m;
      float sn, cs;
      __sincosf(ang, &sn, &cs);
      float2 b = data[i2];
      float2 t = make_float2(cs * b.x - sn * b.y, cs * b.y + sn * b.x);
      float2 a = data[i1];
      data[i1] = make_float2(a.x + t.x, a.y + t.y);
      data[i2] = make_float2(a.x - t.x, a.y - t.y);
    }
    __syncthreads();
  }
}

// Two arrays through the same butterfly schedule: shared twiddles, one
// barrier per stage for both FFTs.
__device__ void fft_lds2(float2* d0, float2* d1, int tid, float sign) {
  for (int s = 1; s <= FFT_LOG; ++s) {
    const int m = 1 << s, half = m >> 1;
    for (int j = tid; j < FFT_N / 2; j += 256) {
      int grp = j >> (s - 1);
      int pos = j & (half - 1);
      int i1 = grp * m + pos;
      int i2 = i1 + half;
      float ang = sign * 6.283185307179586f * (float)pos / (float)m;
      float sn, cs;
      __sincosf(ang, &sn, &cs);
      {
        float2 b = d0[i2];
        float2 t = make_float2(cs * b.x - sn * b.y, cs * b.y + sn * b.x);
        float2 a = d0[i1];
        d0[i1] = make_float2(a.x + t.x, a.y + t.y);
        d0[i2] = make_float2(a.x - t.x, a.y - t.y);
      }
      {
        float2 b = d1[i2];
        float2 t = make_float2(cs * b.x - sn * b.y, cs * b.y + sn * b.x);
        float2 a = d1[i1];
        d1[i1] = make_float2(a.x + t.x, a.y + t.y);
        d1[i2] = make_float2(a.x - t.x, a.y - t.y);
      }
    }
    __syncthreads();
  }
}

// grid (BB, NCHUNK): each block FFTs 32 channels of Q and K, accumulates
// Qf * conj(Kf) into a partial spectrum.
__global__ __launch_bounds__(256) void k_fft_corr(
    const float* __restrict__ Q, const float* __restrict__ Km,
    float2* __restrict__ Spart)
{
  __shared__ float2 bq[FFT_N];
  __shared__ float2 bk[FFT_N];
  __shared__ float2 acc[FFT_N];
  const int b = blockIdx.x, chunk = blockIdx.y, tid = threadIdx.x;
  for (int i = tid; i < FFT_N; i += 256) acc[i] = make_float2(0.f, 0.f);
  const size_t baseb = (size_t)b * LSEQ * DM;
  for (int dd = 0; dd < 32; ++dd) {
    const int d = chunk * 32 + dd;
    __syncthreads();
    for (int i = tid; i < FFT_N; i += 256) {
      int br = __brev((unsigned)i) >> (32 - FFT_LOG);
      bq[br] = make_float2(Q[baseb + (size_t)i * DM + d], 0.f);
      bk[br] = make_float2(Km[baseb + (size_t)i * DM + d], 0.f);
    }
    __syncthreads();
    fft_lds2(bq, bk, tid, -1.f);
    for (int i = tid; i < FFT_N; i += 256) {
      float2 q = bq[i], k = bk[i];
      acc[i].x += q.x * k.x + q.y * k.y;   // q * conj(k)
      acc[i].y += q.y * k.x - q.x * k.y;
    }
  }
  __syncthreads();
  float2* out = Spart + ((size_t)b * NCHUNK + chunk) * FFT_N;
  for (int i = tid; i < FFT_N; i += 256) out[i] = acc[i];
}

// grid BB: reduce partial spectra, inverse FFT, scale by 1/(N*512) -> mv[b,tau]
__global__ __launch_bounds__(256) void k_fft_finish(
    const float2* __restrict__ Spart, float* __restrict__ mv)
{
  __shared__ float2 sum[FFT_N];
  __shared__ float2 data[FFT_N];
  const int b = blockIdx.x, tid = threadIdx.x;
  for (int i = tid; i < FFT_N; i += 256) {
    float2 s = make_float2(0.f, 0.f);
    for (int c = 0; c < NCHUNK; ++c) {
      float2 v = Spart[((size_t)b * NCHUNK + c) * FFT_N + i];
      s.x += v.x; s.y += v.y;
    }
    sum[i] = s;
  }
  __syncthreads();
  for (int i = tid; i < FFT_N; i += 256) {
    int br = __brev((unsigned)i) >> (32 - FFT_LOG);
    data[br] = sum[i];
  }
  __syncthreads();
  fft_lds(data, tid, +1.f);
  const float scale = 1.0f / ((float)FFT_N * 512.0f);
  for (int i = tid; i < FFT_N; i += 256)
    mv[(size_t)b * LSEQ + i] = data[i].x * scale;
}

// single block: g = sum_b mv; 38x masked argmax -> indices
__global__ __launch_bounds__(256) void k_topk(
    const float* __restrict__ mv, int* __restrict__ index)
{
  __shared__ float g[LSEQ];
  __shared__ float smax[256];
  __shared__ int   simax[256];
  const int tid = threadIdx.x;
  for (int i = tid; i < LSEQ; i += 256) {
    float s = 0.f;
    for (int b = 0; b < BB; ++b) s += mv[(size_t)b * LSEQ + i];
    g[i] = s;
  }
  __syncthreads();
  for (int t = 0; t < TOPK; ++t) {
    float best = -INFINITY; int bi = 0x7fffffff;
    for (int i = tid; i < LSEQ; i += 256)
      if (g[i] > best) { best = g[i]; bi = i; }
    smax[tid] = best; simax[tid] = bi;
    __syncthreads();
    for (int off = 128; off > 0; off >>= 1) {
      if (tid < off) {
        if (smax[tid + off] > smax[tid] ||
            (smax[tid + off] == smax[tid] && simax[tid + off] < simax[tid])) {
          smax[tid] = smax[tid + off];
          simax[tid] = simax[tid + off];
        }
      }
      __syncthreads();
    }
    if (tid == 0) { index[t] = simax[0]; g[simax[0]] = -INFINITY; }
    __syncthreads();
  }
}

__global__ void k_softmax_w(const float* __restrict__ mv,
                            const int* __restrict__ index,
                            float* __restrict__ w)
{
  const int b = threadIdx.x;
  if (b >= BB) return;
  float vals[TOPK];
  float mx = -INFINITY;
  for (int i = 0; i < TOPK; ++i) {
    float v = mv[(size_t)b * LSEQ + index[i]];
    vals[i] = v;
    mx = fmaxf(mx, v);
  }
  float s = 0.f;
  for (int i = 0; i < TOPK; ++i) { vals[i] = expf(vals[i] - mx); s += vals[i]; }
  float inv = 1.0f / s;
  for (int i = 0; i < TOPK; ++i) w[b * TOPK + i] = vals[i] * inv;
}

// grid = B*L blocks; out[b,t,d] = sum_i w[b,i] * V[b,(t+idx_i)%L,d]
__global__ __launch_bounds__(256) void k_aggregate(
    const float* __restrict__ V, const float* __restrict__ w,
    const int* __restrict__ index, float* __restrict__ out)
{
  __shared__ int   sidx[TOPK];
  __shared__ float sw[TOPK];
  const int bt = blockIdx.x;
  const int b = bt >> 11, t = bt & (LSEQ - 1);
  const int tid = threadIdx.x;
  if (tid < TOPK) { sidx[tid] = index[tid]; sw[tid] = w[b * TOPK + tid]; }
  __syncthreads();
  const size_t baseb = (size_t)b * LSEQ * DM;
  float a0 = 0.f, a1 = 0.f;
  for (int i = 0; i < TOPK; ++i) {
    int row = (t + sidx[i]) & (LSEQ - 1);
    const float* vp = V + baseb + (size_t)row * DM;
    float wi = sw[i];
    a0 += wi * vp[tid];
    a1 += wi * vp[tid + 256];
  }
  float* op = out + baseb + (size_t)t * DM;
  op[tid] = a0;
  op[tid + 256] = a1;
}

// ============================ glue kernels =================================
// series_decomp: moving average k=25 with replicate padding.
// trend_mode: 0 = skip trend store, 1 = store trend, 2 = accumulate trend.
__global__ void k_decomp(const float* __restrict__ x, float* __restrict__ seasonal,
                         float* __restrict__ trend, int L, int Cn, int trend_mode)
{
  size_t id = (size_t)blockIdx.x * blockDim.x + threadIdx.x;
  size_t total = (size_t)BB * L * Cn;
  if (id >= total) return;
  int c = (int)(id % Cn);
  size_t bt = id / Cn;
  int t = (int)(bt % L);
  int b = (int)(bt / L);
  const float* xb = x + (size_t)b * L * Cn;
  float s = 0.f;
  for (int j = -12; j <= 12; ++j) {
    int tt = t + j;
    tt = tt < 0 ? 0 : (tt >= L ? L - 1 : tt);
    s += xb[(size_t)tt * Cn + c];
  }
  float mov = s * 0.04f; // 1/25
  seasonal[id] = x[id] - mov;
  if (trend_mode == 1) trend[id] = mov;
  else if (trend_mode == 2) trend[id] += mov;
}

// circ_conv3(x[B,L,7], W[3,7,DM]) + temporal embedding from int marks
__global__ void k_embed(const float* __restrict__ x, const int* __restrict__ mark,
                        const float* __restrict__ W, float* __restrict__ out)
{
  size_t id = (size_t)blockIdx.x * blockDim.x + threadIdx.x;
  if (id >= (size_t)BB * LSEQ * DM) return;
  int c = (int)(id & (DM - 1));
  size_t bt = id >> 9;
  int t = (int)(bt & (LSEQ - 1));
  int b = (int)(bt >> 11);
  const float* xb = x + (size_t)b * LSEQ * 7;
  int tp = (t == 0) ? LSEQ - 1 : t - 1;
  int tn = (t == LSEQ - 1) ? 0 : t + 1;
  float s = 0.f;
  for (int j = 0; j < 7; ++j) {
    s += xb[(size_t)tp * 7 + j] * W[(0 * 7 + j) * DM + c];
    s += xb[(size_t)t  * 7 + j] * W[(1 * 7 + j) * DM + c];
    s += xb[(size_t)tn * 7 + j] * W[(2 * 7 + j) * DM + c];
  }
  const int* mk = mark + ((size_t)b * LSEQ + t) * 4;
  int p = c >> 1;
  float div = expf((float)(2 * p) * (-9.210340371976184f / 512.0f));
  for (int j = 0; j < 4; ++j) {
    float a = (float)mk[j] * div;
    s += (c & 1) ? cosf(a) : sinf(a);
  }
  out[id] = s;
}

__global__ void k_meanx(const float* __restrict__ x, float* __restrict__ mean)
{
  int id = threadIdx.x;
  if (id >= BB * 7) return;
  int b = id / 7, c = id % 7;
  float s = 0.f;
  for (int t = 0; t < LSEQ; ++t) s += x[((size_t)b * LSEQ + t) * 7 + c];
  mean[id] = s / (float)LSEQ;
}

// seasonal_init = [seasonal[:, -1024:], zeros]; trend = [trend[:, -1024:], mean]
__global__ void k_inits(const float* __restrict__ seas, const float* __restrict__ tr,
                        const float* __restrict__ mean,
                        float* __restrict__ seasonal_init, float* __restrict__ trend_out)
{
  size_t id = (size_t)blockIdx.x * blockDim.x + threadIdx.x;
  if (id >= (size_t)BB * LSEQ * 7) return;
  int c = (int)(id % 7);
  size_t bt = id / 7;
  int t = (int)(bt % LSEQ);
  int b = (int)(bt / LSEQ);
  if (t < 1024) {
    size_t src = ((size_t)b * LSEQ + 1024 + t) * 7 + c;
    seasonal_init[id] = seas[src];
    trend_out[id] = tr[src];
  } else {
    seasonal_init[id] = 0.f;
    trend_out[id] = mean[b * 7 + c];
  }
}

// layernorm over last dim (block per row)
__global__ __launch_bounds__(128) void k_layernorm(
    const float* __restrict__ x, const float* __restrict__ w,
    const float* __restrict__ bb, float* __restrict__ out)
{
  __shared__ float red[128];
  const int row = blockIdx.x, tid = threadIdx.x;
  const float* xr = x + (size_t)row * DM;
  float v[4], s = 0.f, sq = 0.f;
#pragma unroll
  for (int i = 0; i < 4; ++i) { v[i] = xr[tid + i * 128]; s += v[i]; sq += v[i] * v[i]; }
  red[tid] = s; __syncthreads();
  for (int o = 64; o > 0; o >>= 1) { if (tid < o) red[tid] += red[tid + o]; __syncthreads(); }
  float mu = red[0] / (float)DM; __syncthreads();
  red[tid] = sq; __syncthreads();
  for (int o = 64; o > 0; o >>= 1) { if (tid < o) red[tid] += red[tid + o]; __syncthreads(); }
  float var = red[0] / (float)DM - mu * mu;
  float inv = rsqrtf(var + 1e-5f);
#pragma unroll
  for (int i = 0; i < 4; ++i) {
    int c = tid + i * 128;
    out[(size_t)row * DM + c] = (v[i] - mu) * inv * w[c] + bb[c];
  }
}

__global__ void k_colmean(const float* __restrict__ x, float* __restrict__ cm)
{
  int id = blockIdx.x * blockDim.x + threadIdx.x;
  if (id >= BB * DM) return;
  int b = id >> 9, c = id & (DM - 1);
  const float* xp = x + (size_t)b * LSEQ * DM + c;
  float s = 0.f;
  for (int t = 0; t < LSEQ; ++t) s += xp[(size_t)t * DM];
  cm[id] = s / (float)LSEQ;
}

__global__ void k_submean(float* __restrict__ x, const float* __restrict__ cm)
{
  size_t id = (size_t)blockIdx.x * blockDim.x + threadIdx.x;
  if (id >= (size_t)BB * LSEQ * DM) return;
  int c = (int)(id & (DM - 1));
  int b = (int)(id >> 20);      // / (2048*512)
  x[id] -= cm[b * DM + c];
}

// trend += circ_conv3(S[B,L,DM], W[3,DM,7])
__global__ void k_trendconv(const float* __restrict__ S, const float* __restrict__ W,
                            float* __restrict__ trend)
{
  size_t id = (size_t)blockIdx.x * blockDim.x + threadIdx.x;
  if (id >= (size_t)BB * LSEQ * 7) return;
  int c = (int)(id % 7);
  size_t bt = id / 7;
  int t = (int)(bt % LSEQ);
  int b = (int)(bt / LSEQ);
  int tp = (t == 0) ? LSEQ - 1 : t - 1;
  int tn = (t == LSEQ - 1) ? 0 : t + 1;
  const float* sb = S + (size_t)b * LSEQ * DM;
  float s = 0.f;
  for (int d = 0; d < DM; ++d) {
    s += sb[(size_t)tp * DM + d] * W[(0 * DM + d) * 7 + c]
       + sb[(size_t)t  * DM + d] * W[(1 * DM + d) * 7 + c]
       + sb[(size_t)tn * DM + d] * W[(2 * DM + d) * 7 + c];
  }
  trend[id] += s;
}

// out[b, t-1024, c] = trend[b,t,c] + dec_ln[b,t,:] @ projW[:,c] + projb[c]
__global__ void k_final(const float* __restrict__ dec_ln, const float* __restrict__ projW,
                        const float* __restrict__ projb, const float* __restrict__ trend,
                        float* __restrict__ out)
{
  size_t id = (size_t)blockIdx.x * blockDim.x + threadIdx.x;
  if (id >= (size_t)BB * 1024 * 7) return;
  int c = (int)(id % 7);
  size_t bt = id / 7;
  int tt = (int)(bt % 1024);
  int b = (int)(bt / 1024);
  int t = 1024 + tt;
  const float* xr = dec_ln + ((size_t)b * LSEQ + t) * DM;
  float s = projb[c];
  for (int d = 0; d < DM; ++d) s += xr[d] * projW[d * 7 + c];
  out[id] = s + trend[((size_t)b * LSEQ + t) * 7 + c];
}

// ============================ host orchestration ===========================
extern "C" void kernel_launch(void* const* d_in, const int* in_sizes, int n_in,
                              void* d_out, int out_size, void* d_ws, size_t ws_size,
                              hipStream_t stream)
{
  const float* x_enc      = (const float*)d_in[0];
  const int*   x_mark_enc = (const int*)  d_in[1];
  const int*   x_mark_dec = (const int*)  d_in[3];
  const float* emb_enc_W  = (const float*)d_in[4];
  const float* emb_dec_W  = (const float*)d_in[5];
  const float* enc_Wq = (const float*)d_in[6],  *enc_bq = (const float*)d_in[7];
  const float* dec_sWq= (const float*)d_in[8],  *dec_sbq= (const float*)d_in[9];
  const float* dec_cWq= (const float*)d_in[10], *dec_cbq= (const float*)d_in[11];
  const float* enc_Wk = (const float*)d_in[12], *enc_bk = (const float*)d_in[13];
  const float* dec_sWk= (const float*)d_in[14], *dec_sbk= (const float*)d_in[15];
  const float* dec_cWk= (const float*)d_in[16], *dec_cbk= (const float*)d_in[17];
  const float* enc_Wv = (const float*)d_in[18], *enc_bv = (const float*)d_in[19];
  const float* dec_sWv= (const float*)d_in[20], *dec_sbv= (const float*)d_in[21];
  const float* dec_cWv= (const float*)d_in[22], *dec_cbv= (const float*)d_in[23];
  const float* enc_Wo = (const float*)d_in[24], *enc_bo = (const float*)d_in[25];
  const float* dec_sWo= (const float*)d_in[26], *dec_sbo= (const float*)d_in[27];
  const float* dec_cWo= (const float*)d_in[28], *dec_cbo= (const float*)d_in[29];
  const float* enc_c1 = (const float*)d_in[30], *enc_c2 = (const float*)d_in[31];
  const float* enc_nw = (const float*)d_in[32], *enc_nb = (const float*)d_in[33];
  const float* dec_c1 = (const float*)d_in[34], *dec_c2 = (const float*)d_in[35];
  const float* dec_trendW = (const float*)d_in[36];
  const float* dec_nw = (const float*)d_in[37], *dec_nb = (const float*)d_in[38];
  const float* proj_W = (const float*)d_in[39], *proj_b = (const float*)d_in[40];

  const size_t BIG = (size_t)MTOT * DM;             // 8.4M floats per buffer
  float* base = (float*)d_ws;
  float* bE    = base + 0 * BIG;   // encoder state
  float* bENC  = base + 1 * BIG;   // encoder output after my_layernorm
  float* bD    = base + 2 * BIG;   // decoder state
  float* bQ    = base + 3 * BIG;   // Q / FFN-mid
  float* bK    = base + 4 * BIG;
  float* bV    = base + 5 * BIG;
  float* bA    = base + 6 * BIG;   // aggregation output
  float* bX    = base + 7 * BIG;   // residual-sum / LN scratch
  float* bY    = base + 8 * BIG;   // seasonal after decomp
  float* tsum  = base + 9 * BIG;   // t1+t2+t3 in decoder layer
  float* small = base + 10 * BIG;
  float2* Spart = (float2*)small;                          // 8*16*2048 cplx
  float*  mv    = small + 2 * (size_t)BB * NCHUNK * FFT_N; // B*2048
  float*  cm    = mv + (size_t)BB * LSEQ;                  // B*512
  float*  meanx = cm + BB * DM;                            // 56
  float*  seas7 = meanx + 64;                              // B*L*7
  float*  tr7   = seas7 + (size_t)BB * LSEQ * 7;
  float*  sinit = tr7   + (size_t)BB * LSEQ * 7;
  float*  trend = sinit + (size_t)BB * LSEQ * 7;
  float*  wts   = trend + (size_t)BB * LSEQ * 7;           // B*TOPK
  int*    idx   = (int*)(wts + BB * 64);

  const int ELT_BLK = 256;
  const size_t nBLD = (size_t)MTOT * DM;                   // B*L*512 elems
  const int gBLD = (int)((nBLD + ELT_BLK - 1) / ELT_BLK);  // 32768 blocks
  const size_t nBL7 = (size_t)MTOT * 7;
  const int gBL7 = (int)((nBL7 + ELT_BLK - 1) / ELT_BLK);

  auto gemm = [&](const float* A, const float* W, const float* bias,
                  const float* resid, float* C, int gelu_flag) {
    k_gemm_wmma<<<dim3(DM / BN, MTOT / BM), 256, 0, stream>>>(
        A, W, bias, resid, C, MTOT, DM, DM, gelu_flag);
  };

  auto attention = [&](const float* qx, const float* kx, const float* vx,
                       const float* Wq, const float* bq, const float* Wk, const float* bk,
                       const float* Wv, const float* bv, const float* Wo, const float* bo,
                       const float* residual, float* outbuf) {
    gemm(qx, Wq, bq, nullptr, bQ, 0);
    gemm(kx, Wk, bk, nullptr, bK, 0);
    gemm(vx, Wv, bv, nullptr, bV, 0);
    k_fft_corr<<<dim3(BB, NCHUNK), 256, 0, stream>>>(bQ, bK, Spart);
    k_fft_finish<<<BB, 256, 0, stream>>>(Spart, mv);
    k_topk<<<1, 256, 0, stream>>>(mv, idx);
    k_softmax_w<<<1, 32, 0, stream>>>(mv, idx, wts);
    k_aggregate<<<MTOT, 256, 0, stream>>>(bV, wts, idx, bA);
    gemm(bA, Wo, bo, residual, outbuf, 0);   // residual fused: x + attn(x)
  };

  auto my_layernorm = [&](const float* x, const float* w, const float* b, float* out) {
    k_layernorm<<<MTOT, 128, 0, stream>>>(x, w, b, out);
    k_colmean<<<(BB * DM + 255) / 256, 256, 0, stream>>>(out, cm);
    k_submean<<<gBLD, ELT_BLK, 0, stream>>>(out, cm);
  };

  // ---- init: decomposition of x_enc, seasonal/trend inits -----------------
  k_meanx<<<1, 64, 0, stream>>>(x_enc, meanx);
  k_decomp<<<gBL7, ELT_BLK, 0, stream>>>(x_enc, seas7, tr7, LSEQ, 7, 1);
  k_inits<<<gBL7, ELT_BLK, 0, stream>>>(seas7, tr7, meanx, sinit, trend);

  // ---- encoder embedding + layers -----------------------------------------
  k_embed<<<gBLD, ELT_BLK, 0, stream>>>(x_enc, x_mark_enc, emb_enc_W, bE);
  for (int l = 0; l < 3; ++l) {
    const size_t wo = (size_t)l * DM * DM;
    const size_t bo = (size_t)l * DM;
    attention(bE, bE, bE,
              enc_Wq + wo, enc_bq + bo, enc_Wk + wo, enc_bk + bo,
              enc_Wv + wo, enc_bv + bo, enc_Wo + wo, enc_bo + bo,
              /*residual=*/bE, bX);                         // bX = enc_out + new_x
    k_decomp<<<gBLD, ELT_BLK, 0, stream>>>(bX, bY, nullptr, LSEQ, DM, 0);
    gemm(bY, enc_c1 + wo, nullptr, nullptr, bQ, /*gelu=*/1); // gelu(x @ c1)
    gemm(bQ, enc_c2 + wo, nullptr, /*residual=*/bY, bX, 0);  // x + y
    k_decomp<<<gBLD, ELT_BLK, 0, stream>>>(bX, bE, nullptr, LSEQ, DM, 0);
  }
  my_layernorm(bE, enc_nw, enc_nb, bENC);

  // ---- decoder embedding + layers -----------------------------------------
  k_embed<<<gBLD, ELT_BLK, 0, stream>>>(sinit, x_mark_dec, emb_dec_W, bD);
  for (int l = 0; l < 2; ++l) {
    const size_t wo = (size_t)l * DM * DM;
    const size_t bo = (size_t)l * DM;
    // self attention + residual
    attention(bD, bD, bD,
              dec_sWq + wo, dec_sbq + bo, dec_sWk + wo, dec_sbk + bo,
              dec_sWv + wo, dec_sbv + bo, dec_sWo + wo, dec_sbo + bo,
              bD, bX);
    k_decomp<<<gBLD, ELT_BLK, 0, stream>>>(bX, bY, tsum, LSEQ, DM, 1);   // t1
    // cross attention + residual
    attention(bY, bENC, bENC,
              dec_cWq + wo, dec_cbq + bo, dec_cWk + wo, dec_cbk + bo,
              dec_cWv + wo, dec_cbv + bo, dec_cWo + wo, dec_cbo + bo,
              bY, bX);
    k_decomp<<<gBLD, ELT_BLK, 0, stream>>>(bX, bA, tsum, LSEQ, DM, 2);   // +t2
    // FFN
    gemm(bA, dec_c1 + wo, nullptr, nullptr, bQ, 1);
    gemm(bQ, dec_c2 + wo, nullptr, bA, bX, 0);
    k_decomp<<<gBLD, ELT_BLK, 0, stream>>>(bX, bD, tsum, LSEQ, DM, 2);   // +t3
    // trend += circ_conv3(t1+t2+t3, trendW[l])
    k_trendconv<<<gBL7, ELT_BLK, 0, stream>>>(tsum, dec_trendW + (size_t)l * 3 * DM * 7,
                                              trend);
  }

  // ---- final: layernorm, projection, trend add, slice ---------------------
  my_layernorm(bD, dec_nw, dec_nb, bX);
  const size_t nout = (size_t)BB * 1024 * 7;
  k_final<<<(int)((nout + 255) / 256), 256, 0, stream>>>(bX, proj_W, proj_b,
                                                         trend, (float*)d_out);
}